// ISSDiagonalizer_22024592294286
// MI455X (gfx1250) — compile-verified
//
#include <hip/hip_runtime.h>

typedef __attribute__((ext_vector_type(2))) float v2f;
typedef __attribute__((ext_vector_type(4))) float v4f;
typedef __attribute__((ext_vector_type(8))) float v8f;

#define Bb 16
#define Ff 257
#define Mm 6
#define Tt 1000
#define BF (Bb * Ff)
#define XST 1012      // mult of 4 (b128 align); 1012%64=52 -> 16 rows hit 16 distinct banks
#define EPSv 1e-6f

__device__ __forceinline__ v8f wmma_f32_16x16x4(v2f a, v2f b, v8f c) {
  // 8-arg form: (neg_a, A, neg_b, B, c_mod, C, reuse_a, reuse_b)
  return __builtin_amdgcn_wmma_f32_16x16x4_f32(false, a, false, b, (short)0, c,
                                               false, false);
}

// ---- Sequential ISS k-loop over the 6 rows (runs on first 36 threads) ----
__device__ void iss_kloop(float* Vre, float* Vim, float* Qre, float* Qim,
                          float* qre, float* qim, float* Vqre, float* Vqim,
                          float* vre, float* vim, int tid) {
  for (int ks = 0; ks < 6; ++ks) {
    __syncthreads();
    if (tid < 6) { qre[tid] = Qre[ks * 6 + tid]; qim[tid] = Qim[ks * 6 + tid]; }
    __syncthreads();
    if (tid < 36) {  // Vq[kk][m] = sum_n V[kk][m][n] * conj(q[n])
      int kk = tid / 6, m = tid % 6;
      float sre = 0.f, sim = 0.f;
#pragma unroll
      for (int n = 0; n < 6; ++n) {
        float vr = Vre[(kk * 6 + m) * 6 + n];
        float vi = Vim[(kk * 6 + m) * 6 + n];
        sre += vr * qre[n] + vi * qim[n];
        sim += vi * qre[n] - vr * qim[n];
      }
      Vqre[tid] = sre; Vqim[tid] = sim;
    }
    __syncthreads();
    if (tid < 6) {
      int kk = tid;
      float qq = 0.f, nre = 0.f, nim = 0.f;
#pragma unroll
      for (int m = 0; m < 6; ++m) {
        float vqr = Vqre[kk * 6 + m], vqi = Vqim[kk * 6 + m];
        qq  += qre[m] * vqr - qim[m] * vqi;       // Re(q . Vq)
        nre += Qre[kk * 6 + m] * vqr - Qim[kk * 6 + m] * vqi;
        nim += Qre[kk * 6 + m] * vqi + Qim[kk * 6 + m] * vqr;
      }
      qq = fmaxf(qq, 1e-6f);
      float vr_, vi_;
      if (kk == ks) { vr_ = 1.0f - rsqrtf(qq); vi_ = 0.0f; }
      else          { vr_ = nre / qq;          vi_ = nim / qq; }
      vre[kk] = vr_; vim[kk] = vi_;
    }
    __syncthreads();
    if (tid < 36) {  // Q -= v outer q
      int kk = tid / 6, m = tid % 6;
      Qre[tid] -= vre[kk] * qre[m] - vim[kk] * qim[m];
      Qim[tid] -= vre[kk] * qim[m] + vim[kk] * qre[m];
    }
  }
  __syncthreads();
}

// ---- Kernel 1: WMMA covariance V + iteration-1 k-loop + xt partial sums ----
__global__ __launch_bounds__(128) void k1_cov_iter1(
    const float* __restrict__ r, const float* __restrict__ Qre_in,
    const float* __restrict__ Qim_in, const float* __restrict__ xre,
    const float* __restrict__ xim, float* __restrict__ wsV,
    float* __restrict__ wsQ, float* __restrict__ wsPart) {
  __shared__ float xs[16 * XST];     // rows 0-5 = x_re, 6-11 = x_im, 12-15 = zero pad
  __shared__ float rs[6 * Tt];
  __shared__ float red[4 * 6 * 256]; // per-wave full 16x16 C per k, stored [n][m]
  __shared__ float Vre[216], Vim[216];
  __shared__ float Qre[36], Qim[36];
  __shared__ float qre[6], qim[6], Vqre[36], Vqim[36], vre[6], vim[6];
  __shared__ float sh[128];

  const int tid = threadIdx.x;
  const int bf = blockIdx.x;
  const float* rb  = r   + (size_t)bf * 6000;
  const float* xrb = xre + (size_t)bf * 6000;
  const float* xib = xim + (size_t)bf * 6000;

  // Stage r and x with b128 transactions; zero pad rows 12-15
  for (int row = 0; row < 6; ++row) {
    const v4f* sr = (const v4f*)(xrb + row * Tt);
    const v4f* si = (const v4f*)(xib + row * Tt);
    const v4f* sw = (const v4f*)(rb + row * Tt);
    for (int t4 = tid; t4 < Tt / 4; t4 += 128) {
      *(v4f*)&xs[row * XST + 4 * t4]       = sr[t4];
      *(v4f*)&xs[(row + 6) * XST + 4 * t4] = si[t4];
      *(v4f*)&rs[row * Tt + 4 * t4]        = sw[t4];
    }
  }
  {
    const v4f z4 = {0.f, 0.f, 0.f, 0.f};
    for (int i = tid; i < XST; i += 128) *(v4f*)&xs[12 * XST + 4 * i] = z4;
  }
  __syncthreads();

  // WMMA: per k, C_k = (r_k .* [xr;xi]) @ [xr;xi]^T  (12x12 in a 16x16 tile)
  const int lane = tid & 31;
  const int wv   = tid >> 5;
  const int half = lane >> 4;   // K half: lanes 16-31 hold K=2,3
  const int rowi = lane & 15;   // M / N index
  const float* xp = &xs[rowi * XST];
  v8f acc[6];
#pragma unroll
  for (int k = 0; k < 6; ++k) acc[k] = (v8f){0, 0, 0, 0, 0, 0, 0, 0};

  for (int t0 = wv * 4; t0 < Tt; t0 += 16) {
    const int tA = t0 + 2 * half;      // VGPR j holds K = 2*half + j  ->  t = tA + j
    v2f xf;
    xf[0] = xp[tA];
    xf[1] = xp[tA + 1];
#pragma unroll
    for (int k = 0; k < 6; ++k) {
      const float r0 = rs[k * Tt + tA];
      const float r1 = rs[k * Tt + tA + 1];
      v2f af; af[0] = r0 * xf[0]; af[1] = r1 * xf[1];
      acc[k] = wmma_f32_16x16x4(af, xf, acc[k]);
    }
  }

  // Dump full tiles transposed: C(m,n) -> red[(wv*6+k)*256 + n*16 + m]
  // (lane: n=rowi fixed, m = vgpr + 8*half contiguous -> two b128 stores per k)
#pragma unroll
  for (int k = 0; k < 6; ++k) {
    float* base = &red[(wv * 6 + k) * 256 + rowi * 16 + 8 * half];
    v4f lo = {acc[k][0], acc[k][1], acc[k][2], acc[k][3]};
    v4f hi = {acc[k][4], acc[k][5], acc[k][6], acc[k][7]};
    *(v4f*)(base)     = lo;
    *(v4f*)(base + 4) = hi;
  }
  __syncthreads();

  // Reduce 4 waves, fold complex blocks, scale by 1/T, add EPS*I; spill V to ws
  const float invT = 1.0f / (float)Tt;
  for (int idx = tid; idx < 216; idx += 128) {
    const int k = idx / 36, rem = idx % 36, m = rem / 6, n = rem % 6;
    float sre = 0.f, sim = 0.f;
#pragma unroll
    for (int w = 0; w < 4; ++w) {
      const float* cb = &red[(w * 6 + k) * 256];  // C(m,n) at cb[n*16 + m]
      sre += cb[n * 16 + m] + cb[(n + 6) * 16 + (m + 6)];
      sim += cb[n * 16 + (m + 6)] - cb[(n + 6) * 16 + m];
    }
    sre *= invT; sim *= invT;
    if (m == n) sre += EPSv;
    Vre[idx] = sre; Vim[idx] = sim;
    wsV[(size_t)bf * 432 + idx] = sre;
    wsV[(size_t)bf * 432 + 216 + idx] = sim;
  }
  if (tid < 36) {
    Qre[tid] = Qre_in[(size_t)bf * 36 + tid];
    Qim[tid] = Qim_in[(size_t)bf * 36 + tid];
  }
  __syncthreads();

  iss_kloop(Vre, Vim, Qre, Qim, qre, qim, Vqre, Vqim, vre, vim, tid);

  if (tid < 36) {  // Q after iter-1 k-loop (pre-scale)
    wsQ[(size_t)bf * 72 + tid]      = Qre[tid];
    wsQ[(size_t)bf * 72 + 36 + tid] = Qim[tid];
  }

  // xt partial sum for scale1, reusing staged x (b128 LDS reads, 4 t at a time)
  float lsum = 0.f;
  for (int t4 = tid; t4 < Tt / 4; t4 += 128) {
    v4f xr_[6], xi_[6];
#pragma unroll
    for (int n = 0; n < 6; ++n) {
      xr_[n] = *(const v4f*)&xs[n * XST + 4 * t4];
      xi_[n] = *(const v4f*)&xs[(n + 6) * XST + 4 * t4];
    }
#pragma unroll
    for (int m = 0; m < 6; ++m) {
      v4f ar = {0.f, 0.f, 0.f, 0.f}, ai = {0.f, 0.f, 0.f, 0.f};
#pragma unroll
      for (int n = 0; n < 6; ++n) {
        const float qr = Qre[m * 6 + n], qi = Qim[m * 6 + n];
        ar += qr * xr_[n] - qi * xi_[n];
        ai += qr * xi_[n] + qi * xr_[n];
      }
      v4f p = ar * ar + ai * ai;
      lsum += p[0] + p[1] + p[2] + p[3];
    }
  }
  sh[tid] = lsum; __syncthreads();
  for (int s = 64; s > 0; s >>= 1) { if (tid < s) sh[tid] += sh[tid + s]; __syncthreads(); }
  if (tid == 0) wsPart[bf] = sh[0];
}

// ---- per-batch scale after iteration 1: Q /= sqrt(clip(scale)) ----
__global__ __launch_bounds__(256) void scale1_k(const float* __restrict__ wsPart,
                                                float* __restrict__ wsQ) {
  __shared__ float sh[256];
  __shared__ float shqs;
  const int b = blockIdx.x, tid = threadIdx.x;
  float v = wsPart[b * Ff + tid];
  if (tid == 0) v += wsPart[b * Ff + 256];
  sh[tid] = v; __syncthreads();
  for (int s = 128; s > 0; s >>= 1) { if (tid < s) sh[tid] += sh[tid + s]; __syncthreads(); }
  if (tid == 0) {
    const float scale = sh[0] / (float)(Ff * Mm * Tt);
    shqs = rsqrtf(fmaxf(scale, 1e-6f));
  }
  __syncthreads();
  const float qs = shqs;
  for (int i = tid; i < Ff * 72; i += 256) wsQ[(size_t)b * Ff * 72 + i] *= qs;
}

// ---- Kernel 2: iteration-2 k-loop + xt (unnormalized) + partial sums ----
__global__ __launch_bounds__(128) void k2_iter2(
    const float* __restrict__ xre, const float* __restrict__ xim,
    const float* __restrict__ wsV, float* __restrict__ wsQ,
    float* __restrict__ wsPart, float* __restrict__ xt_out) {
  __shared__ float Vre[216], Vim[216], Qre[36], Qim[36];
  __shared__ float qre[6], qim[6], Vqre[36], Vqim[36], vre[6], vim[6];
  __shared__ float sh[128];
  const int tid = threadIdx.x;
  const int bf = blockIdx.x;
  for (int i = tid; i < 216; i += 128) {
    Vre[i] = wsV[(size_t)bf * 432 + i];
    Vim[i] = wsV[(size_t)bf * 432 + 216 + i];
  }
  if (tid < 36) {
    Qre[tid] = wsQ[(size_t)bf * 72 + tid];
    Qim[tid] = wsQ[(size_t)bf * 72 + 36 + tid];
  }
  __syncthreads();

  iss_kloop(Vre, Vim, Qre, Qim, qre, qim, Vqre, Vqim, vre, vim, tid);

  if (tid < 36) {
    wsQ[(size_t)bf * 72 + tid]      = Qre[tid];
    wsQ[(size_t)bf * 72 + 36 + tid] = Qim[tid];
  }

  const float* xrb = xre + (size_t)bf * 6000;
  const float* xib = xim + (size_t)bf * 6000;
  float* xtb = xt_out + (size_t)bf * 6000;
  float lsum = 0.f;
  for (int t4 = tid; t4 < Tt / 4; t4 += 128) {
    v4f xr_[6], xi_[6];
#pragma unroll
    for (int n = 0; n < 6; ++n) {
      xr_[n] = *(const v4f*)(xrb + n * Tt + 4 * t4);
      xi_[n] = *(const v4f*)(xib + n * Tt + 4 * t4);
    }
#pragma unroll
    for (int m = 0; m < 6; ++m) {
      v4f ar = {0.f, 0.f, 0.f, 0.f}, ai = {0.f, 0.f, 0.f, 0.f};
#pragma unroll
      for (int n = 0; n < 6; ++n) {
        const float qr = Qre[m * 6 + n], qi = Qim[m * 6 + n];
        ar += qr * xr_[n] - qi * xi_[n];
        ai += qr * xi_[n] + qi * xr_[n];
      }
      v4f p = ar * ar + ai * ai;
      *(v4f*)(xtb + m * Tt + 4 * t4) = p;  // unnormalized; rescaled by xtnorm_k
      lsum += p[0] + p[1] + p[2] + p[3];
    }
  }
  sh[tid] = lsum; __syncthreads();
  for (int s = 64; s > 0; s >>= 1) { if (tid < s) sh[tid] += sh[tid + s]; __syncthreads(); }
  if (tid == 0) wsPart[bf] = sh[0];
}

// ---- scale after iteration 2: final Q (complex interleaved) + inv scale ----
__global__ __launch_bounds__(256) void scale2_k(const float* __restrict__ wsPart,
                                                const float* __restrict__ wsQ,
                                                float* __restrict__ wsInv,
                                                float* __restrict__ q_out) {
  __shared__ float sh[256];
  __shared__ float shqs;
  const int b = blockIdx.x, tid = threadIdx.x;
  float v = wsPart[b * Ff + tid];
  if (tid == 0) v += wsPart[b * Ff + 256];
  sh[tid] = v; __syncthreads();
  for (int s = 128; s > 0; s >>= 1) { if (tid < s) sh[tid] += sh[tid + s]; __syncthreads(); }
  if (tid == 0) {
    const float scale = sh[0] / (float)(Ff * Mm * Tt);
    shqs = rsqrtf(fmaxf(scale, 1e-6f));
    wsInv[b] = 1.0f / scale;  // raw scale for xt division (matches reference)
  }
  __syncthreads();
  const float qs = shqs;
  for (int i = tid; i < Ff * 36; i += 256) {
    const int f = i / 36, e = i % 36;
    const size_t bfq = (size_t)(b * Ff + f);
    q_out[2 * (bfq * 36 + e)]     = wsQ[bfq * 72 + e] * qs;
    q_out[2 * (bfq * 36 + e) + 1] = wsQ[bfq * 72 + 36 + e] * qs;
  }
}

// ---- xt /= scale(b), in place over d_out ----
__global__ __launch_bounds__(256) void xtnorm_k(float* __restrict__ xt,
                                                const float* __restrict__ wsInv) {
  const size_t N = (size_t)Bb * Ff * Mm * Tt;
  for (size_t i = (size_t)blockIdx.x * 256 + threadIdx.x; i < N;
       i += (size_t)gridDim.x * 256) {
    const int b = (int)(i / ((size_t)Ff * Mm * Tt));
    xt[i] *= wsInv[b];
  }
}

extern "C" void kernel_launch(void* const* d_in, const int* in_sizes, int n_in,
                              void* d_out, int out_size, void* d_ws, size_t ws_size,
                              hipStream_t stream) {
  (void)in_sizes; (void)n_in; (void)out_size; (void)ws_size;
  const float* r    = (const float*)d_in[0];
  const float* Qre  = (const float*)d_in[1];
  const float* Qim  = (const float*)d_in[2];
  const float* xre  = (const float*)d_in[3];
  const float* xim  = (const float*)d_in[4];

  float* out    = (float*)d_out;
  float* q_out  = out;                            // B*F*36 complex interleaved
  float* xt_out = out + (size_t)BF * 36 * 2;      // B*F*M*T floats

  float* ws     = (float*)d_ws;
  float* wsV    = ws;                             // BF*432 floats
  float* wsQ    = wsV + (size_t)BF * 432;         // BF*72 floats
  float* wsPart = wsQ + (size_t)BF * 72;          // BF floats
  float* wsInv  = wsPart + BF;                    // Bb floats

  k1_cov_iter1<<<dim3(BF), dim3(128), 0, stream>>>(r, Qre, Qim, xre, xim, wsV, wsQ, wsPart);
  scale1_k<<<dim3(Bb), dim3(256), 0, stream>>>(wsPart, wsQ);
  k2_iter2<<<dim3(BF), dim3(128), 0, stream>>>(xre, xim, wsV, wsQ, wsPart, xt_out);
  scale2_k<<<dim3(Bb), dim3(256), 0, stream>>>(wsPart, wsQ, wsInv, q_out);
  xtnorm_k<<<dim3(2048), dim3(256), 0, stream>>>(xt_out, wsInv);
}